// TabNetAttentiveTransformer_26259430048013
// MI455X (gfx1250) — compile-verified
//
#include <hip/hip_runtime.h>
#include <stdint.h>

typedef float v2f __attribute__((ext_vector_type(2)));
typedef float v8f __attribute__((ext_vector_type(8)));
typedef unsigned int u32x4 __attribute__((ext_vector_type(4)));
typedef int i32x4 __attribute__((ext_vector_type(4)));
typedef int i32x8 __attribute__((ext_vector_type(8)));

#define VBS     64
#define DDIM    256
#define IN_DIM  512
#define KCHUNK  64
#define NCHUNK  (IN_DIM / KCHUNK)   // 8
#define BN_EPS  1e-3f

// LDS layout (float offsets). Double-buffered staging; sZ aliases buffer 0.
#define SX0_OFF 0
#define SW0_OFF 4096
#define SX1_OFF 20480
#define SW1_OFF 24576
#define SMEM_FLOATS 40960           // 160 KB

// Build a 2D TDM descriptor (D#) and issue TENSOR_LOAD_TO_LDS.
// Group0: count=1 | lds_addr | global_addr(57b) | type=2.
// Group1: data_size=4B, tensor_dim0/1, tile_dim0/1, tensor_dim0_stride.
// Groups 2/3/4 zero (<=2D tensor). 6-arg builtin form (clang-23).
__device__ __forceinline__ void tdm_load_2d(uint32_t lds_byte_off,
                                            const void* gaddr,
                                            uint32_t tensor_d0, uint32_t tensor_d1,
                                            uint32_t tile_d0,   uint32_t tile_d1,
                                            uint32_t stride0_elems)
{
    uint64_t ga = (uint64_t)(uintptr_t)gaddr;
    u32x4 g0;
    g0[0] = 1u;                                                  // count=1 (user D#)
    g0[1] = lds_byte_off;                                        // lds_addr (bytes)
    g0[2] = (uint32_t)(ga & 0xFFFFFFFFu);                        // global_addr[31:0]
    g0[3] = (uint32_t)((ga >> 32) & 0x01FFFFFFu) | (2u << 30);   // addr[56:32] | type=2

    i32x8 g1;
    g1[0] = (int)(2u << 16);                                     // data_size=2 (4 bytes)
    g1[1] = (int)((tensor_d0 & 0xFFFFu) << 16);                  // tensor_dim0[15:0]
    g1[2] = (int)(((tensor_d0 >> 16) & 0xFFFFu) |
                  ((tensor_d1 & 0xFFFFu) << 16));                // dim0[31:16] | dim1[15:0]
    g1[3] = (int)(((tensor_d1 >> 16) & 0xFFFFu) |
                  ((tile_d0 & 0xFFFFu) << 16));                  // dim1[31:16] | tile_dim0
    g1[4] = (int)(tile_d1 & 0xFFFFu);                            // tile_dim1 (tile_dim2=0)
    g1[5] = (int)stride0_elems;                                  // tensor_dim0_stride[31:0]
    g1[6] = 0;                                                   // stride hi / dim1_stride lo
    g1[7] = 0;

    i32x4 g2 = {0, 0, 0, 0};
    i32x4 g3 = {0, 0, 0, 0};
    i32x8 g4 = {0, 0, 0, 0, 0, 0, 0, 0};
    __builtin_amdgcn_tensor_load_to_lds(g0, g1, g2, g3, g4, 0);
}

__launch_bounds__(256)
__global__ void tabnet_attn_kernel(const float* __restrict__ X,
                                   const float* __restrict__ prior,
                                   const float* __restrict__ W,
                                   const float* __restrict__ gamma,
                                   const float* __restrict__ beta,
                                   float* __restrict__ out)
{
    __shared__ float smem[SMEM_FLOATS];
    float* sZ = smem;                 // 64x256, aliases buffer 0 after GEMM

    const int tid  = threadIdx.x;
    const int lane = tid & 31;
    const int wave = tid >> 5;
    const int row0 = blockIdx.x * VBS;

    // 2 row-tiles x 4 col-tiles per wave: fewer LDS fragment loads per WMMA.
    const int rq    = wave >> 2;         // row-tile pair: rows tiles {2rq, 2rq+1}
    const int cq    = wave & 3;          // col-tile quad: col tiles {4cq .. 4cq+3}
    const int l15   = lane & 15;         // M for A, N for B/C
    const int kb    = (lane >> 4) << 1;  // K sub-offset: 0 or 2 (ISA fp32 A/B layout)

    const uint32_t lds_base = (uint32_t)(uintptr_t)(void*)smem;  // LDS byte offset

    v8f acc[2][4] = {};

    // Issue chunk 0 (X tile: 64 rows x 64 k, strided; W chunk: 64x256 contiguous).
    if (wave == 0) {
        tdm_load_2d(lds_base + SX0_OFF * 4, X + (size_t)row0 * IN_DIM,
                    KCHUNK, VBS, KCHUNK, VBS, IN_DIM);
        tdm_load_2d(lds_base + SW0_OFF * 4, W,
                    KCHUNK * DDIM, 1, KCHUNK * DDIM, 1, KCHUNK * DDIM);
    }

    for (int c = 0; c < NCHUNK; ++c) {
        if (wave == 0) {
            if (c + 1 < NCHUNK) {
                const int kk = (c + 1) * KCHUNK;
                const uint32_t bx = ((c + 1) & 1) ? SX1_OFF : SX0_OFF;
                const uint32_t bw = ((c + 1) & 1) ? SW1_OFF : SW0_OFF;
                tdm_load_2d(lds_base + bx * 4, X + (size_t)row0 * IN_DIM + kk,
                            KCHUNK, VBS, KCHUNK, VBS, IN_DIM);
                tdm_load_2d(lds_base + bw * 4, W + (size_t)kk * DDIM,
                            KCHUNK * DDIM, 1, KCHUNK * DDIM, 1, KCHUNK * DDIM);
                __builtin_amdgcn_s_wait_tensorcnt(2);  // chunk c landed (in-order)
            } else {
                __builtin_amdgcn_s_wait_tensorcnt(0);  // everything landed
            }
        }
        __syncthreads();

        const float* sX = smem + ((c & 1) ? SX1_OFF : SX0_OFF);
        const float* sW = smem + ((c & 1) ? SW1_OFF : SW0_OFF);

        for (int k4 = 0; k4 < KCHUNK; k4 += 4) {
            // Two A fragments (row tiles 2rq, 2rq+1): contiguous b64 in LDS.
            v2f a0, a1;
            {
                const float* ax0 = &sX[((rq * 2 + 0) * 16 + l15) * KCHUNK + k4 + kb];
                const float* ax1 = &sX[((rq * 2 + 1) * 16 + l15) * KCHUNK + k4 + kb];
                a0.x = ax0[0]; a0.y = ax0[1];
                a1.x = ax1[0]; a1.y = ax1[1];
            }
            #pragma unroll
            for (int cc = 0; cc < 4; ++cc) {
                // B fragment 4x16: lane -> (K=k4+kb / +1, N=l15); reused by 2 WMMAs.
                v2f b;
                b.x = sW[(k4 + kb    ) * DDIM + (cq * 4 + cc) * 16 + l15];
                b.y = sW[(k4 + kb + 1) * DDIM + (cq * 4 + cc) * 16 + l15];
                acc[0][cc] = __builtin_amdgcn_wmma_f32_16x16x4_f32(
                    false, a0, false, b, (short)0, acc[0][cc], false, false);
                acc[1][cc] = __builtin_amdgcn_wmma_f32_16x16x4_f32(
                    false, a1, false, b, (short)0, acc[1][cc], false, false);
            }
        }
        __syncthreads();
    }

    // Spill accumulators to sZ (C/D layout: VGPR r -> M = r + 8*(lane>=16), N = lane&15).
    const int Mc = (lane >> 4) * 8;
    #pragma unroll
    for (int rt2 = 0; rt2 < 2; ++rt2) {
        const int rt = rq * 2 + rt2;
        #pragma unroll
        for (int cc = 0; cc < 4; ++cc) {
            #pragma unroll
            for (int r = 0; r < 8; ++r) {
                sZ[(rt * 16 + Mc + r) * DDIM + (cq * 4 + cc) * 16 + l15] = acc[rt2][cc][r];
            }
        }
    }
    __syncthreads();

    // Phase 2: virtual-batch BatchNorm (stats over 64 rows per column) + prior scale.
    {
        const int d = tid;
        float mean = 0.f;
        for (int r = 0; r < VBS; ++r) mean += sZ[r * DDIM + d];
        mean *= (1.0f / VBS);
        float var = 0.f;
        for (int r = 0; r < VBS; ++r) {
            float t = sZ[r * DDIM + d] - mean;
            var += t * t;
        }
        var *= (1.0f / VBS);
        float inv = rsqrtf(var + BN_EPS) * gamma[d];
        float bet = beta[d];
        for (int r = 0; r < VBS; ++r) {
            float zn = (sZ[r * DDIM + d] - mean) * inv + bet;
            zn *= prior[(size_t)(row0 + r) * DDIM + d];
            sZ[r * DDIM + d] = zn;
        }
    }
    __syncthreads();

    // Phase 3: sparsemax per row (Michelot fixed point, exact on convergence).
    for (int r8 = 0; r8 < 8; ++r8) {
        const int r = wave * 8 + r8;
        float zv[8];
        #pragma unroll
        for (int j = 0; j < 8; ++j) zv[j] = sZ[r * DDIM + lane + 32 * j];

        float s = 0.f;
        #pragma unroll
        for (int j = 0; j < 8; ++j) s += zv[j];
        #pragma unroll
        for (int off = 16; off > 0; off >>= 1) s += __shfl_xor(s, off, 32);

        int   k   = DDIM;
        float tau = (s - 1.0f) / (float)k;
        for (int iter = 0; iter < DDIM; ++iter) {
            float s2 = 0.f;
            int   k2 = 0;
            #pragma unroll
            for (int j = 0; j < 8; ++j) {
                if (zv[j] > tau) { s2 += zv[j]; k2++; }
            }
            #pragma unroll
            for (int off = 16; off > 0; off >>= 1) {
                s2 += __shfl_xor(s2, off, 32);
                k2 += __shfl_xor(k2, off, 32);
            }
            if (k2 == k) break;   // support unchanged -> tau final (wave-uniform)
            k   = k2;
            tau = (s2 - 1.0f) / (float)k2;
        }

        #pragma unroll
        for (int j = 0; j < 8; ++j) {
            out[(size_t)(row0 + r) * DDIM + lane + 32 * j] = fmaxf(zv[j] - tau, 0.0f);
        }
    }
}

extern "C" void kernel_launch(void* const* d_in, const int* in_sizes, int n_in,
                              void* d_out, int out_size, void* d_ws, size_t ws_size,
                              hipStream_t stream) {
    const float* X     = (const float*)d_in[0]; // [B, 512]
    const float* prior = (const float*)d_in[1]; // [B, 256]
    const float* W     = (const float*)d_in[2]; // [512, 256]
    const float* gamma = (const float*)d_in[3]; // [256]
    const float* beta  = (const float*)d_in[4]; // [256]
    float* out = (float*)d_out;                 // [B, 256]

    const int B = in_sizes[0] / IN_DIM;         // 131072
    const int nblocks = B / VBS;                // 2048 virtual batches
    tabnet_attn_kernel<<<nblocks, 256, 0, stream>>>(X, prior, W, gamma, beta, out);
}